// GraphAttentionLayer_time_serial_83665962926396
// MI455X (gfx1250) — compile-verified
//
#include <hip/hip_runtime.h>
#include <hip/hip_bf16.h>
#include <math.h>

#define NTIME 20
#define V 128
#define F 128
#define ALPHA 0.2f
#define NEGMASK -9.0e15f

typedef float v2f __attribute__((ext_vector_type(2)));
typedef float v8f __attribute__((ext_vector_type(8)));

// D = A(16x4) * B(4x16) + C, all f32  ->  v_wmma_f32_16x16x4_f32
__device__ __forceinline__ v8f wmma4(v2f a, v2f b, v8f c) {
    return __builtin_amdgcn_wmma_f32_16x16x4_f32(
        /*neg_a=*/false, a, /*neg_b=*/false, b,
        /*c_mod=*/(short)0, c, /*reuse_a=*/false, /*reuse_b=*/false);
}

// ---------------------------------------------------------------------------
// K1: Wh = h (2560x128) @ W (128x128), fp32 WMMA, 1 wave per 16x16 tile.
// ---------------------------------------------------------------------------
__global__ void __launch_bounds__(128)
wh_gemm_kernel(const float* __restrict__ h, const float* __restrict__ W,
               float* __restrict__ Wh) {
    const int lane = threadIdx.x & 31;
    const int wave = threadIdx.x >> 5;
    const int tile = blockIdx.x * 4 + wave;   // 1280 tiles total
    const int mi   = tile >> 3;               // 0..159  (row tile)
    const int ni   = tile & 7;                // 0..7    (col tile)
    const int mn   = lane & 15;               // A-row / B-col index
    const int kh   = (lane >> 4) << 1;        // K sub-offset: 0 or 2

    const float* arow = h + (size_t)(mi * 16 + mn) * F;
    const float* bcol = W + ni * 16 + mn;

    v8f c = {};
#pragma unroll
    for (int k0 = 0; k0 < F; k0 += 4) {
        v2f a = *(const v2f*)(arow + k0 + kh);                 // A[m][k..k+1]
        v2f b;
        b.x = bcol[(size_t)(k0 + kh) * F];                     // B[k][n]
        b.y = bcol[(size_t)(k0 + kh + 1) * F];
        c = wmma4(a, b, c);
    }
    // C/D layout: M = v + (lane>>4)*8, N = lane&15
    float* crow = Wh + (size_t)(mi * 16 + kh * 4) * F + ni * 16 + mn;
#pragma unroll
    for (int v = 0; v < 8; ++v) crow[(size_t)v * F] = c[v];
}

// ---------------------------------------------------------------------------
// K2: e1 = Wh @ a1, e2 = Wh @ a2 (a1 = a[0:128], a2 = a[128:256]).
// Shared A-operand, two WMMA accumulators.
// ---------------------------------------------------------------------------
__global__ void __launch_bounds__(128)
e12_gemm_kernel(const float* __restrict__ Wh, const float* __restrict__ a,
                float* __restrict__ e1, float* __restrict__ e2) {
    const int lane = threadIdx.x & 31;
    const int wave = threadIdx.x >> 5;
    const int tile = blockIdx.x * 4 + wave;
    const int mi   = tile >> 3;
    const int ni   = tile & 7;
    const int mn   = lane & 15;
    const int kh   = (lane >> 4) << 1;

    const float* arow = Wh + (size_t)(mi * 16 + mn) * F;
    const float* b1   = a + ni * 16 + mn;                 // a1[k][n]
    const float* b2   = a + (size_t)F * F + ni * 16 + mn; // a2[k][n]

    v8f c1 = {}, c2 = {};
#pragma unroll
    for (int k0 = 0; k0 < F; k0 += 4) {
        v2f av = *(const v2f*)(arow + k0 + kh);
        v2f bv1, bv2;
        bv1.x = b1[(size_t)(k0 + kh) * F];
        bv1.y = b1[(size_t)(k0 + kh + 1) * F];
        bv2.x = b2[(size_t)(k0 + kh) * F];
        bv2.y = b2[(size_t)(k0 + kh + 1) * F];
        c1 = wmma4(av, bv1, c1);
        c2 = wmma4(av, bv2, c2);
    }
    float* r1 = e1 + (size_t)(mi * 16 + kh * 4) * F + ni * 16 + mn;
    float* r2 = e2 + (size_t)(mi * 16 + kh * 4) * F + ni * 16 + mn;
#pragma unroll
    for (int v = 0; v < 8; ++v) { r1[(size_t)v * F] = c1[v]; r2[(size_t)v * F] = c2[v]; }
}

// ---------------------------------------------------------------------------
// K3: per (t,b) block, thread = f. Online softmax over j, fused attention
// store (non-temporal: 168 MB write-once stream) + weighted sum + ELU.
// ---------------------------------------------------------------------------
__global__ void __launch_bounds__(128)
attn_softmax_kernel(const float* __restrict__ e1, const float* __restrict__ e2,
                    const float* __restrict__ Wh, const float* __restrict__ adj,
                    float* __restrict__ out, float* __restrict__ att) {
    const int tb = blockIdx.x;          // t*V + b, 0..2559
    const int t  = tb >> 7;
    const int b  = tb & (V - 1);
    const int f  = threadIdx.x;         // 0..127

    const float e1v   = e1[(size_t)tb * F + f];
    const float* e2t  = e2  + (size_t)t * V * F;
    const float* Wht  = Wh  + (size_t)t * V * F;

    __shared__ float s_adj[V];
    s_adj[f] = adj[(size_t)b * V + f];
    __syncthreads();

    // pass 1: online max + sum over j
    float m = -INFINITY, s = 0.f;
    for (int j = 0; j < V; ++j) {
        float e = e1v + e2t[(size_t)j * F + f];
        e = e > 0.f ? e : ALPHA * e;                 // LeakyReLU
        float val = s_adj[j] > 0.f ? e : NEGMASK;    // adjacency mask
        float nm = fmaxf(m, val);
        s = s * __expf(m - nm) + __expf(val - nm);
        m = nm;
    }
    const float inv = 1.0f / s;

    // pass 2: normalize, stream attention out, accumulate h_prime
    float acc = 0.f;
    float* attb = att + (size_t)tb * V * F + f;
    for (int j = 0; j < V; ++j) {
        float e = e1v + e2t[(size_t)j * F + f];
        e = e > 0.f ? e : ALPHA * e;
        float val = s_adj[j] > 0.f ? e : NEGMASK;
        float av = __expf(val - m) * inv;
        __builtin_nontemporal_store(av, attb + (size_t)j * F);
        acc = fmaf(av, Wht[(size_t)j * F + f], acc);
    }
    out[(size_t)tb * F + f] = acc > 0.f ? acc : (__expf(acc) - 1.f);  // ELU
}

// ---------------------------------------------------------------------------
extern "C" void kernel_launch(void* const* d_in, const int* in_sizes, int n_in,
                              void* d_out, int out_size, void* d_ws, size_t ws_size,
                              hipStream_t stream) {
    const float* h   = (const float*)d_in[0];   // (NTIME*V, F)
    const float* adj = (const float*)d_in[1];   // (V, V, 1)
    const float* W   = (const float*)d_in[2];   // (F, F)
    const float* a   = (const float*)d_in[3];   // (2F, F)

    float* out = (float*)d_out;                         // (NTIME, V, F)
    float* att = out + (size_t)NTIME * V * F;           // (NTIME, V, V, F)

    float* Wh = (float*)d_ws;                           // (NTIME*V, F)
    float* e1 = Wh + (size_t)NTIME * V * F;
    float* e2 = e1 + (size_t)NTIME * V * F;

    const int tiles = (NTIME * V / 16) * (F / 16);      // 1280 wave-tiles

    wh_gemm_kernel     <<<tiles / 4, 128, 0, stream>>>(h, W, Wh);
    e12_gemm_kernel    <<<tiles / 4, 128, 0, stream>>>(Wh, a, e1, e2);
    attn_softmax_kernel<<<NTIME * V, 128, 0, stream>>>(e1, e2, Wh, adj, out, att);
}